// SoftmaxLinearAttention_51634096832893
// MI455X (gfx1250) — compile-verified
//
#include <hip/hip_runtime.h>
#include <hip/hip_bf16.h>

// Problem constants (match reference)
#define BDIM   2
#define TDIM   2048
#define DMODEL 1024
#define NHEADS 16
#define HSZ    64
#define CHUNK  32

typedef __attribute__((ext_vector_type(2))) float v2f;
typedef __attribute__((ext_vector_type(8))) float v8f;
typedef __attribute__((ext_vector_type(4))) unsigned int u32x4;
typedef __attribute__((ext_vector_type(8))) int i32x8;
typedef __attribute__((ext_vector_type(4))) int i32x4;

// ---------------------------------------------------------------------------
// WMMA helpers: V_WMMA_F32_16X16X4_F32, D(16x16 f32) = A(16x4) * B(4x16) + C
//   A frag: lane L, vgpr v -> A[L%16][2*(L/16)+v]
//   B frag: lane L, vgpr v -> B[2*(L/16)+v][L%16]
//   C/D:    lane L, vgpr r -> C[r + 8*(L/16)][L%16]
// ---------------------------------------------------------------------------
static __device__ __forceinline__ v8f wmma_f32_k4(v2f a, v2f b, v8f c) {
  return __builtin_amdgcn_wmma_f32_16x16x4_f32(
      false, a, false, b, (short)0, c, false, false);
}

static __device__ __forceinline__ v2f frag_mk(const float* P, int ld, int m0, int kk, int lane) {
  const int r = m0 + (lane & 15);
  const int k = kk + ((lane >> 4) << 1);
  v2f f; f.x = P[r * ld + k]; f.y = P[r * ld + k + 1];
  return f;
}

static __device__ __forceinline__ v2f frag_kn(const float* P, int ld, int kk, int n0, int lane) {
  const int k = kk + ((lane >> 4) << 1);
  const int n = n0 + (lane & 15);
  v2f f; f.x = P[k * ld + n]; f.y = P[(k + 1) * ld + n];
  return f;
}

static __device__ __forceinline__ void store_cd(float* P, int ld, int m0, int n0, int lane, v8f acc) {
  const int mb = m0 + ((lane >> 4) << 3);
  const int n  = n0 + (lane & 15);
#pragma unroll
  for (int r = 0; r < 8; ++r) P[(mb + r) * ld + n] = acc[r];
}

static __device__ __forceinline__ v8f load_cd(const float* P, int ld, int m0, int n0, int lane) {
  const int mb = m0 + ((lane >> 4) << 3);
  const int n  = n0 + (lane & 15);
  v8f v;
#pragma unroll
  for (int r = 0; r < 8; ++r) v[r] = P[(mb + r) * ld + n];
  return v;
}

// ---------------------------------------------------------------------------
// TDM: 2D tile load Global -> LDS (fp32 elements), with LDS row padding.
// D# per CDNA5 ISA. pad_icode: 3 -> every 16 dwords, 5 -> every 64.
// pad amount code 0 == 1 dword (reproduces [rows][w+1] LDS layout).
// Generic->LDS address: low 32 bits of a generic LDS pointer are the LDS offset.
// ---------------------------------------------------------------------------
static __device__ __forceinline__ unsigned lds_off(const void* p) {
  return (unsigned)(unsigned long long)(size_t)p;
}

static __device__ __forceinline__ void tdm_load_2d(
    unsigned lds_addr, const float* gptr,
    unsigned tile_w, unsigned tile_h,          // in dwords / rows
    unsigned tensor_w, unsigned tensor_h,      // tensor extents (dwords / rows)
    unsigned row_stride,                       // dwords between rows
    unsigned pad_icode) {
  const unsigned long long ga = (unsigned long long)(size_t)gptr;
  u32x4 g0;
  g0.x = 1u;                                        // count=1, user descriptor
  g0.y = lds_addr;                                  // lds_addr [63:32]
  g0.z = (unsigned)(ga & 0xffffffffu);              // global_addr lo
  g0.w = (unsigned)((ga >> 32) & 0x01ffffffu) | (2u << 30);  // addr hi | type=2
  i32x8 g1;
  g1[0] = (int)((2u << 16) |                        // data_size = 4B
                (1u << 20) |                        // pad_enable
                (pad_icode << 22));                 // pad_interval; pad_amount=0 (1 dword)
  g1[1] = (int)((tensor_w & 0xffffu) << 16);        // tensor_dim0[15:0]
  g1[2] = (int)(((tensor_w >> 16) & 0xffffu) | ((tensor_h & 0xffffu) << 16));
  g1[3] = (int)(((tensor_h >> 16) & 0xffffu) | ((tile_w & 0xffffu) << 16));
  g1[4] = (int)(tile_h & 0xffffu);                  // tile_dim1; tile_dim2=0
  g1[5] = (int)row_stride;                          // tensor_dim0_stride[31:0]
  g1[6] = 0;
  g1[7] = 0;
  const i32x4 z4 = {};
  const i32x8 z8 = {};
  __builtin_amdgcn_tensor_load_to_lds(g0, g1, z4, z4, z8, 0);
}

// ---------------------------------------------------------------------------
// Kernel 1: qkv = x @ W_qkv^T  (M=4096, N=3072, K=1024), fp32 WMMA.
// Block tile 128x64, BK=16, TDM double-buffered staging, 8 waves x (2x2) subtiles.
// Scattered store into q/k/v buffers laid out [b][h][t][c].
// ---------------------------------------------------------------------------
__global__ __launch_bounds__(256) void qkv_gemm_kernel(
    const float* __restrict__ X, const float* __restrict__ Wqkv,
    float* __restrict__ qb, float* __restrict__ kb, float* __restrict__ vb) {
  __shared__ float Xs[2][128][17];
  __shared__ float Ws[2][64][17];
  const int tid  = threadIdx.x;
  const int lane = tid & 31, wave = tid >> 5;
  const int row0 = blockIdx.y * 128, col0 = blockIdx.x * 64;
  const int mi0 = (wave & 3) * 2, ni0 = (wave >> 2) * 2;
  const int NST = DMODEL / 16;

  v8f a00 = {}, a01 = {}, a10 = {}, a11 = {};

  if (tid == 0) {
    tdm_load_2d(lds_off(&Xs[0][0][0]), X + (size_t)row0 * DMODEL,
                16, 128, DMODEL, BDIM * TDIM, DMODEL, 3);
    tdm_load_2d(lds_off(&Ws[0][0][0]), Wqkv + (size_t)col0 * DMODEL,
                16, 64, DMODEL, 3 * DMODEL, DMODEL, 3);
  }

  for (int s = 0; s < NST; ++s) {
    const int cur = s & 1;
    if (tid == 0) {
      if (s + 1 < NST) {
        const int k1 = (s + 1) * 16;
        tdm_load_2d(lds_off(&Xs[cur ^ 1][0][0]), X + (size_t)row0 * DMODEL + k1,
                    16, 128, DMODEL, BDIM * TDIM, DMODEL, 3);
        tdm_load_2d(lds_off(&Ws[cur ^ 1][0][0]), Wqkv + (size_t)col0 * DMODEL + k1,
                    16, 64, DMODEL, 3 * DMODEL, DMODEL, 3);
        __builtin_amdgcn_s_wait_tensorcnt(2);   // current buffer complete
      } else {
        __builtin_amdgcn_s_wait_tensorcnt(0);
      }
    }
    __syncthreads();
#pragma unroll
    for (int kk = 0; kk < 16; kk += 4) {
      const v2f fa0 = frag_mk(&Xs[cur][0][0], 17, (mi0 + 0) * 16, kk, lane);
      const v2f fa1 = frag_mk(&Xs[cur][0][0], 17, (mi0 + 1) * 16, kk, lane);
      const v2f fb0 = frag_mk(&Ws[cur][0][0], 17, (ni0 + 0) * 16, kk, lane);
      const v2f fb1 = frag_mk(&Ws[cur][0][0], 17, (ni0 + 1) * 16, kk, lane);
      a00 = wmma_f32_k4(fa0, fb0, a00);
      a01 = wmma_f32_k4(fa0, fb1, a01);
      a10 = wmma_f32_k4(fa1, fb0, a10);
      a11 = wmma_f32_k4(fa1, fb1, a11);
    }
    __syncthreads();
  }

  // Scatter: col j -> (three=j/1024, h=(j%1024)/64, c=j%64); row bt -> (b,t)
#pragma unroll
  for (int im = 0; im < 2; ++im) {
#pragma unroll
    for (int in = 0; in < 2; ++in) {
      const v8f acc = im ? (in ? a11 : a10) : (in ? a01 : a00);
      const int j = col0 + (ni0 + in) * 16 + (lane & 15);
      const int three = j >> 10, rem = j & 1023, h = rem >> 6, c = rem & 63;
      float* dst = (three == 0) ? qb : ((three == 1) ? kb : vb);
      const int mb = row0 + (mi0 + im) * 16 + ((lane >> 4) << 3);
#pragma unroll
      for (int r = 0; r < 8; ++r) {
        const int bt = mb + r;
        const int b = bt >> 11, t = bt & (TDIM - 1);
        dst[(((size_t)(b * NHEADS + h)) * TDIM + t) * HSZ + c] = acc[r];
      }
    }
  }
}

// ---------------------------------------------------------------------------
// Kernel 2: kmax[b][h][c] = max over t of k[b][h][t][c]
// ---------------------------------------------------------------------------
__global__ __launch_bounds__(64) void kmax_kernel(const float* __restrict__ kb,
                                                  float* __restrict__ kmax) {
  const int bh = blockIdx.x, c = threadIdx.x;
  const float* p = kb + (size_t)bh * TDIM * HSZ + c;
  float m = -1e30f;
  for (int t = 0; t < TDIM; ++t) m = fmaxf(m, p[(size_t)t * HSZ]);
  kmax[bh * HSZ + c] = m;
}

// ---------------------------------------------------------------------------
// Kernel 3: chunked linear-attention scan, one workgroup per (b,h).
// Per chunk (C=32): TDM-load q/k/v tiles, softmax(q), exp(k-kmax),
// z-cumsum folded into Q', then
//   A = tril(Q' Kexp^T);  out = A @ V + Q' @ S;  S += Kexp^T @ V
// all with fp32 WMMA on LDS tiles.
// ---------------------------------------------------------------------------
__global__ __launch_bounds__(256) void recur_kernel(
    const float* __restrict__ qb, const float* __restrict__ kb,
    const float* __restrict__ vb, const float* __restrict__ kmax,
    float* __restrict__ xo) {
  __shared__ float Sst[64][65];   // state S[c][d]
  __shared__ float Qc[CHUNK][65]; // q -> softmax -> q'
  __shared__ float Kc[CHUNK][65]; // k -> k_exp
  __shared__ float Vc[CHUNK][65];
  __shared__ float Am[CHUNK][33]; // masked Q'Kexp^T
  __shared__ float zprev[64];

  const int tid = threadIdx.x, lane = tid & 31, wave = tid >> 5;
  const int bh = blockIdx.x;
  const int b = bh >> 4, h = bh & 15;
  const size_t base = (size_t)bh * TDIM * HSZ;

  for (int i = tid; i < 64 * 64; i += 256) Sst[i >> 6][i & 63] = 0.f;
  if (tid < 64) zprev[tid] = 0.f;
  __syncthreads();

  for (int ch = 0; ch < TDIM / CHUNK; ++ch) {
    const int t0 = ch * CHUNK;
    // TDM: stage the three 32x64 chunk tiles into padded LDS rows (stride 65)
    if (tid == 0) {
      tdm_load_2d(lds_off(&Qc[0][0]), qb + base + (size_t)t0 * HSZ,
                  HSZ, CHUNK, HSZ, TDIM, HSZ, 5);
      tdm_load_2d(lds_off(&Kc[0][0]), kb + base + (size_t)t0 * HSZ,
                  HSZ, CHUNK, HSZ, TDIM, HSZ, 5);
      tdm_load_2d(lds_off(&Vc[0][0]), vb + base + (size_t)t0 * HSZ,
                  HSZ, CHUNK, HSZ, TDIM, HSZ, 5);
      __builtin_amdgcn_s_wait_tensorcnt(0);
    }
    __syncthreads();
    // k_exp in place
    for (int i = tid; i < CHUNK * HSZ; i += 256) {
      const int t = i >> 6, c = i & 63;
      Kc[t][c] = __expf(Kc[t][c] - kmax[bh * HSZ + c]);
    }
    __syncthreads();
    // softmax over channels, * HS^-0.5 (=0.125)
    if (tid < CHUNK) {
      const int t = tid;
      float m = -1e30f;
      for (int c = 0; c < HSZ; ++c) m = fmaxf(m, Qc[t][c]);
      float s = 0.f;
      for (int c = 0; c < HSZ; ++c) { const float e = __expf(Qc[t][c] - m); Qc[t][c] = e; s += e; }
      const float inv = 0.125f / s;
      for (int c = 0; c < HSZ; ++c) Qc[t][c] *= inv;
    }
    __syncthreads();
    // fold z cumsum into Q'
    if (tid < HSZ) {
      const int c = tid;
      float run = zprev[c];
      for (int t = 0; t < CHUNK; ++t) {
        run += Kc[t][c];
        Qc[t][c] = Qc[t][c] / (run + 1e-9f);
      }
      zprev[c] = run;
    }
    __syncthreads();
    // A = Q' @ Kexp^T (32x32, K=64); waves 0..3
    if (wave < 4) {
      const int mi = wave >> 1, ni = wave & 1;
      v8f acc = {};
#pragma unroll
      for (int kk = 0; kk < HSZ; kk += 4) {
        const v2f a  = frag_mk(&Qc[0][0], 65, mi * 16, kk, lane);
        const v2f bf = frag_mk(&Kc[0][0], 65, ni * 16, kk, lane);
        acc = wmma_f32_k4(a, bf, acc);
      }
      store_cd(&Am[0][0], 33, mi * 16, ni * 16, lane, acc);
    }
    __syncthreads();
    // causal mask (inclusive diagonal)
    for (int i = tid; i < CHUNK * CHUNK; i += 256) {
      const int t = i >> 5, s = i & 31;
      if (s > t) Am[t][s] = 0.f;
    }
    __syncthreads();
    // out = Q' @ S + A @ V (32x64); 8 waves, one 16x16 subtile each
    {
      const int mi = wave >> 2, ni = wave & 3;
      v8f acc = {};
#pragma unroll
      for (int kk = 0; kk < HSZ; kk += 4) {
        const v2f a  = frag_mk(&Qc[0][0], 65, mi * 16, kk, lane);
        const v2f bf = frag_kn(&Sst[0][0], 65, kk, ni * 16, lane);
        acc = wmma_f32_k4(a, bf, acc);
      }
#pragma unroll
      for (int kk = 0; kk < CHUNK; kk += 4) {
        const v2f a  = frag_mk(&Am[0][0], 33, mi * 16, kk, lane);
        const v2f bf = frag_kn(&Vc[0][0], 65, kk, ni * 16, lane);
        acc = wmma_f32_k4(a, bf, acc);
      }
      const int mb = mi * 16 + ((lane >> 4) << 3);
      const int d  = ni * 16 + (lane & 15);
#pragma unroll
      for (int r = 0; r < 8; ++r) {
        const int t = t0 + mb + r;
        xo[((size_t)(b * TDIM + t)) * DMODEL + h * HSZ + d] = acc[r];
      }
    }
    __syncthreads();
    // S += Kexp^T @ V (64x64, K=32); 2 subtiles per wave
#pragma unroll
    for (int si = 0; si < 2; ++si) {
      const int sidx = wave * 2 + si;
      const int mi = sidx >> 2, ni = sidx & 3;
      v8f acc = load_cd(&Sst[0][0], 65, mi * 16, ni * 16, lane);
#pragma unroll
      for (int kk = 0; kk < CHUNK; kk += 4) {
        const v2f a  = frag_kn(&Kc[0][0], 65, kk, mi * 16, lane); // A[c][t]=Kexp[t][c]
        const v2f bf = frag_kn(&Vc[0][0], 65, kk, ni * 16, lane);
        acc = wmma_f32_k4(a, bf, acc);
      }
      store_cd(&Sst[0][0], 65, mi * 16, ni * 16, lane, acc);
    }
    __syncthreads();
  }
}

// ---------------------------------------------------------------------------
// Kernel 4: out = xo @ W_out^T  (M=4096, N=1024, K=1024), fp32 WMMA,
// same 128x64 TDM double-buffered scheme, direct store.
// ---------------------------------------------------------------------------
__global__ __launch_bounds__(256) void out_gemm_kernel(
    const float* __restrict__ X, const float* __restrict__ Wout,
    float* __restrict__ out) {
  __shared__ float Xs[2][128][17];
  __shared__ float Ws[2][64][17];
  const int tid  = threadIdx.x;
  const int lane = tid & 31, wave = tid >> 5;
  const int row0 = blockIdx.y * 128, col0 = blockIdx.x * 64;
  const int mi0 = (wave & 3) * 2, ni0 = (wave >> 2) * 2;
  const int NST = DMODEL / 16;

  v8f a00 = {}, a01 = {}, a10 = {}, a11 = {};

  if (tid == 0) {
    tdm_load_2d(lds_off(&Xs[0][0][0]), X + (size_t)row0 * DMODEL,
                16, 128, DMODEL, BDIM * TDIM, DMODEL, 3);
    tdm_load_2d(lds_off(&Ws[0][0][0]), Wout + (size_t)col0 * DMODEL,
                16, 64, DMODEL, DMODEL, DMODEL, 3);
  }

  for (int s = 0; s < NST; ++s) {
    const int cur = s & 1;
    if (tid == 0) {
      if (s + 1 < NST) {
        const int k1 = (s + 1) * 16;
        tdm_load_2d(lds_off(&Xs[cur ^ 1][0][0]), X + (size_t)row0 * DMODEL + k1,
                    16, 128, DMODEL, BDIM * TDIM, DMODEL, 3);
        tdm_load_2d(lds_off(&Ws[cur ^ 1][0][0]), Wout + (size_t)col0 * DMODEL + k1,
                    16, 64, DMODEL, DMODEL, DMODEL, 3);
        __builtin_amdgcn_s_wait_tensorcnt(2);
      } else {
        __builtin_amdgcn_s_wait_tensorcnt(0);
      }
    }
    __syncthreads();
#pragma unroll
    for (int kk = 0; kk < 16; kk += 4) {
      const v2f fa0 = frag_mk(&Xs[cur][0][0], 17, (mi0 + 0) * 16, kk, lane);
      const v2f fa1 = frag_mk(&Xs[cur][0][0], 17, (mi0 + 1) * 16, kk, lane);
      const v2f fb0 = frag_mk(&Ws[cur][0][0], 17, (ni0 + 0) * 16, kk, lane);
      const v2f fb1 = frag_mk(&Ws[cur][0][0], 17, (ni0 + 1) * 16, kk, lane);
      a00 = wmma_f32_k4(fa0, fb0, a00);
      a01 = wmma_f32_k4(fa0, fb1, a01);
      a10 = wmma_f32_k4(fa1, fb0, a10);
      a11 = wmma_f32_k4(fa1, fb1, a11);
    }
    __syncthreads();
  }

#pragma unroll
  for (int im = 0; im < 2; ++im) {
#pragma unroll
    for (int in = 0; in < 2; ++in) {
      const v8f acc = im ? (in ? a11 : a10) : (in ? a01 : a00);
      const int n  = col0 + (ni0 + in) * 16 + (lane & 15);
      const int mb = row0 + (mi0 + im) * 16 + ((lane >> 4) << 3);
#pragma unroll
      for (int r = 0; r < 8; ++r)
        out[(size_t)(mb + r) * DMODEL + n] = acc[r];
    }
  }
}

// ---------------------------------------------------------------------------
extern "C" void kernel_launch(void* const* d_in, const int* in_sizes, int n_in,
                              void* d_out, int out_size, void* d_ws, size_t ws_size,
                              hipStream_t stream) {
  const float* x    = (const float*)d_in[0];  // (B, T, D)
  const float* Wqkv = (const float*)d_in[1];  // (3D, D)
  const float* Wout = (const float*)d_in[2];  // (D, D)
  float* out = (float*)d_out;                 // (B, T, D)

  float* ws = (float*)d_ws;
  const size_t per = (size_t)BDIM * NHEADS * TDIM * HSZ;  // 4M floats
  float* qb  = ws;
  float* kb  = ws + per;
  float* vb  = ws + 2 * per;
  float* xo  = ws + 3 * per;
  float* kmx = ws + 4 * per;  // 2048 floats

  qkv_gemm_kernel<<<dim3(3 * DMODEL / 64, BDIM * TDIM / 128), 256, 0, stream>>>(x, Wqkv, qb, kb, vb);
  kmax_kernel<<<dim3(BDIM * NHEADS), 64, 0, stream>>>(kb, kmx);
  recur_kernel<<<dim3(BDIM * NHEADS), 256, 0, stream>>>(qb, kb, vb, kmx, xo);
  out_gemm_kernel<<<dim3(DMODEL / 64, BDIM * TDIM / 128), 256, 0, stream>>>(xo, Wout, out);
}